// FormanRicciRegularizer_83356725281084
// MI455X (gfx1250) — compile-verified
//
#include <hip/hip_runtime.h>
#include <hip/hip_bf16.h>
#include <math.h>

typedef float v2f __attribute__((ext_vector_type(2)));
typedef float v8f __attribute__((ext_vector_type(8)));

#define DIM   256
#define KTOP  9        // K_NEIGHBORS + 1
#define NEG_INF (-1e30f)

__device__ __forceinline__ void lds_fence() {
    // wave-local LDS store->load ordering (split counters on gfx1250)
    asm volatile("s_wait_dscnt 0" ::: "memory");
}

// ---------------------------------------------------------------------------
// Kernel 1: row normalization. One wave32 per row (D=256 -> 8 floats/lane).
// ---------------------------------------------------------------------------
__global__ __launch_bounds__(256) void fr_normalize_kernel(
    const float* __restrict__ x, float* __restrict__ xn, int N) {
    int row  = (int)((blockIdx.x * 256 + threadIdx.x) >> 5);
    int lane = threadIdx.x & 31;
    if (row >= N) return;
    const float* rp = x + (size_t)row * DIM;
    float xv[8];
    float ss = 0.f;
#pragma unroll
    for (int q = 0; q < 8; ++q) {
        xv[q] = rp[lane + 32 * q];
        ss += xv[q] * xv[q];
    }
#pragma unroll
    for (int m = 16; m > 0; m >>= 1) ss += __shfl_xor(ss, m, 32);
    float inv = 1.0f / fmaxf(sqrtf(ss), 1e-12f);
    float* op = xn + (size_t)row * DIM;
#pragma unroll
    for (int q = 0; q < 8; ++q) op[lane + 32 * q] = xv[q] * inv;
}

// ---------------------------------------------------------------------------
// Kernel: zero the column counters.
// ---------------------------------------------------------------------------
__global__ void fr_zero_kernel(unsigned* __restrict__ p, int n) {
    int i = blockIdx.x * blockDim.x + threadIdx.x;
    if (i < n) p[i] = 0u;
}

// ---------------------------------------------------------------------------
// 16x16 sim tile via V_WMMA_F32_16X16X4_F32 over K=256.
// Two independent accumulation chains (even/odd K-chunks) for WMMA ILP;
// result written (padded) to the wave's private LDS tile.
// ---------------------------------------------------------------------------
__device__ __forceinline__ void fr_compute_tile(
    const float* __restrict__ brow, const v2f* __restrict__ areg,
    float* __restrict__ mytile, int mn, int rbase) {
    v8f acc0 = {0.f, 0.f, 0.f, 0.f, 0.f, 0.f, 0.f, 0.f};
    v8f acc1 = {0.f, 0.f, 0.f, 0.f, 0.f, 0.f, 0.f, 0.f};
#pragma unroll
    for (int kc = 0; kc < 64; kc += 2) {
        v2f b0 = *(const v2f*)(brow + kc * 4);
        v2f b1 = *(const v2f*)(brow + kc * 4 + 4);
        acc0 = __builtin_amdgcn_wmma_f32_16x16x4_f32(
            false, areg[kc], false, b0, (short)0, acc0, false, false);
        acc1 = __builtin_amdgcn_wmma_f32_16x16x4_f32(
            false, areg[kc + 1], false, b1, (short)0, acc1, false, false);
    }
    // C layout: VGPR v -> row v (lanes 0-15) / v+8 (lanes 16-31), col = mn
#pragma unroll
    for (int v = 0; v < 8; ++v)
        mytile[(rbase + v) * 17 + mn] = acc0[v] + acc1[v];
    lds_fence();
}

// ---------------------------------------------------------------------------
// Fused kernel: block = 128 threads = 4 waves, owns one 16-row strip.
// Phase 1: sweep all column tiles, streaming per-row top-9 -> thresholds
//          (kept in LDS; merged across the 4 waves).
// Phase 2: sweep again, count adjacency row sums r_i (lanes 0-15) and
//          column sums c_j (lanes 16-31, atomics), diagonal excluded.
// A panel (16 rows x K=256) stays pinned in 128 VGPRs across both phases.
// ---------------------------------------------------------------------------
__global__ __launch_bounds__(128) void fr_main_kernel(
    const float* __restrict__ xn, unsigned* __restrict__ rcnt,
    unsigned* __restrict__ ccnt, int N) {
    __shared__ float tile[4][16 * 17];          // per-wave 16x16 tile, padded
    __shared__ float topm[4][16][KTOP];         // per-wave top-9 lists
    __shared__ float thr_s[16];                 // strip thresholds
    __shared__ unsigned rpart[4][16];           // per-wave row-count partials

    const int tid   = threadIdx.x;
    const int lane  = tid & 31;
    const int w     = tid >> 5;
    const int mn    = lane & 15;                // M (A) / N (B) index
    const int kk    = (lane >> 4) << 1;         // k sub-pair: 0 or 2
    const int rbase = (lane >> 4) << 3;         // C row base: 0 or 8
    const int row0  = blockIdx.x * 16;
    const int ntile = N >> 4;
    float* mytile = tile[w];

    // Preload A-panel fragments: rows row0..row0+15, all K=256 -> 64 x v2f
    v2f areg[64];
    const float* arow = xn + (size_t)(row0 + mn) * DIM + kk;
#pragma unroll
    for (int kc = 0; kc < 64; ++kc)
        areg[kc] = *(const v2f*)(arow + kc * 4);

    // ---------------- Phase 1: per-row 9th-largest (thresholds) -----------
    float t[KTOP];
#pragma unroll
    for (int s = 0; s < KTOP; ++s) t[s] = NEG_INF;

    for (int jt = w; jt < ntile; jt += 4) {
        const float* brow = xn + (size_t)(jt * 16 + mn) * DIM + kk;
        fr_compute_tile(brow, areg, mytile, mn, rbase);
        // Every lane merges row (lane&15); lanes 16-31 duplicate (harmless).
#pragma unroll
        for (int c = 0; c < 16; ++c) {
            float vv = mytile[mn * 17 + c];
            if (vv > t[KTOP - 1]) {
                t[KTOP - 1] = vv;
#pragma unroll
                for (int s = KTOP - 1; s > 0; --s) {
                    if (t[s] > t[s - 1]) {
                        float tmp = t[s - 1]; t[s - 1] = t[s]; t[s] = tmp;
                    }
                }
            }
        }
    }

    if (lane < 16) {
#pragma unroll
        for (int s = 0; s < KTOP; ++s) topm[w][lane][s] = t[s];
    }
    __syncthreads();
    if (tid < 16) {
        float vals[4 * KTOP];
#pragma unroll
        for (int ww = 0; ww < 4; ++ww)
#pragma unroll
            for (int s = 0; s < KTOP; ++s)
                vals[ww * KTOP + s] = topm[ww][tid][s];
        float cur = NEG_INF;
#pragma unroll
        for (int it = 0; it < KTOP; ++it) {
            int best = 0;
            cur = vals[0];
#pragma unroll
            for (int s = 1; s < 4 * KTOP; ++s)
                if (vals[s] > cur) { cur = vals[s]; best = s; }
            vals[best] = NEG_INF;
        }
        thr_s[tid] = cur;   // 9th largest (incl. self-similarity)
    }
    __syncthreads();

    // ---------------- Phase 2: adjacency row / column counts --------------
    const float myth = thr_s[mn];
    unsigned rc = 0;

    for (int jt = w; jt < ntile; jt += 4) {
        const float* brow = xn + (size_t)(jt * 16 + mn) * DIM + kk;
        fr_compute_tile(brow, areg, mytile, mn, rbase);

        const int jcol0 = jt * 16;
        const bool diagTile = (jcol0 == row0);
        if (lane < 16) {
            // row mn: count sim >= thr_row, skip diagonal element
#pragma unroll
            for (int c = 0; c < 16; ++c) {
                float vv = mytile[mn * 17 + c];
                rc += (vv >= myth && !(diagTile && c == mn)) ? 1u : 0u;
            }
        } else {
            // column mn: count sim[r][mn] >= thr[r], skip diagonal element
            unsigned cc = 0;
#pragma unroll
            for (int r = 0; r < 16; ++r) {
                float vv = mytile[r * 17 + mn];
                cc += (vv >= thr_s[r] && !(diagTile && r == mn)) ? 1u : 0u;
            }
            atomicAdd(&ccnt[jcol0 + mn], cc);
        }
    }

    if (lane < 16) rpart[w][mn] = rc;
    __syncthreads();
    if (tid < 16)
        rcnt[row0 + tid] = rpart[0][tid] + rpart[1][tid] + rpart[2][tid] + rpart[3][tid];
}

// ---------------------------------------------------------------------------
// Final reduction: loss = (λ/N²) Σ_i [ (r_i + c_i)·max(r_i,1) − 2·r_i ]
// (relu is always pass-through because deg >= 1)
// ---------------------------------------------------------------------------
__global__ __launch_bounds__(256) void fr_reduce_kernel(
    const unsigned* __restrict__ rcnt, const unsigned* __restrict__ ccnt,
    float* __restrict__ out, int N) {
    __shared__ float red[256];
    float s = 0.f;
    for (int i = threadIdx.x; i < N; i += 256) {
        float r = (float)rcnt[i];
        float c = (float)ccnt[i];
        float d = fmaxf(r, 1.0f);
        s += (r + c) * d - 2.0f * r;
    }
    red[threadIdx.x] = s;
    __syncthreads();
    for (int off = 128; off > 0; off >>= 1) {
        if (threadIdx.x < off) red[threadIdx.x] += red[threadIdx.x + off];
        __syncthreads();
    }
    if (threadIdx.x == 0) {
        float nn = (float)N * (float)N;
        out[0] = red[0] * (0.01f / nn);
    }
}

// ---------------------------------------------------------------------------
extern "C" void kernel_launch(void* const* d_in, const int* in_sizes, int n_in,
                              void* d_out, int out_size, void* d_ws, size_t ws_size,
                              hipStream_t stream) {
    const float* emb = (const float*)d_in[0];
    const int N = in_sizes[0] / DIM;          // 12288

    float*    xn   = (float*)d_ws;            // N*256 floats
    unsigned* rcnt = (unsigned*)(xn + (size_t)N * DIM);  // N u32
    unsigned* ccnt = rcnt + N;                            // N u32
    float*    out  = (float*)d_out;

    fr_normalize_kernel<<<(N * 32 + 255) / 256, 256, 0, stream>>>(emb, xn, N);
    fr_zero_kernel<<<(N + 255) / 256, 256, 0, stream>>>(ccnt, N);
    fr_main_kernel<<<N / 16, 128, 0, stream>>>(xn, rcnt, ccnt, N);
    fr_reduce_kernel<<<1, 256, 0, stream>>>(rcnt, ccnt, out, N);
}